// SSDDecodeLayer_57466662420953
// MI455X (gfx1250) — compile-verified
//
#include <hip/hip_runtime.h>
#include <cstdint>

// ---------------- problem constants (match reference) ----------------
#define B_IMG    64
#define N_ANCH   8732
#define N_CH     33                 // 21 conf + 4 loc + 4 anchor + 4 variance
#define N_CLS    21
#define N_USED   (N_CLS - 1)        // classes 1..20 participate
#define NMS_MAX  100
#define TOP_K    200
#define ROWS_PER_IMG (N_USED * NMS_MAX)   // 2000 candidate rows / image
#define BLK      256                // 8 waves (wave32)
#define KPT      ((N_ANCH + BLK - 1) / BLK)   // 35 anchors per thread
#define NEGF     (-1e30f)
#define CONF_TH  0.5f
#define IOU_TH   0.45f
#define IMG_SC   300.0f

// ---------------- CDNA5 TDM (Tensor Data Mover) ----------------
// Pack a D# per CDNA5 ISA and issue
// tensor_load_to_lds: one DMA moves the whole 139.7KB box tile into LDS.
// This toolchain exposes the 6-arg builtin:
//   (uint32x4 g0, int32x8 g1, int32x4 g2, int32x4 g3, int32x8 extra, i32 cpol)
#if defined(__HIP_DEVICE_COMPILE__) && __has_builtin(__builtin_amdgcn_tensor_load_to_lds) && __has_builtin(__builtin_amdgcn_s_wait_tensorcnt)
#define HAVE_TDM 1
#else
#define HAVE_TDM 0
#endif

#if HAVE_TDM
typedef __attribute__((ext_vector_type(4))) unsigned int v4u;
typedef __attribute__((ext_vector_type(8))) int          v8i;
typedef __attribute__((ext_vector_type(4))) int          v4i;

// 2D f32 tile load: tile_d0 elements per row, tile_d1 rows, row stride in elems.
__device__ __forceinline__ void tdm_load_f32(unsigned lds_byte_off, const void* gptr,
                                             unsigned tensor_d0, unsigned tensor_d1,
                                             unsigned tile_d0,   unsigned tile_d1,
                                             unsigned long long stride0_elems) {
  unsigned long long ga = (unsigned long long)gptr;
  v4u g0;
  g0.x = 1u;                                            // count=1, user descriptor
  g0.y = lds_byte_off;                                  // lds_addr (bytes)
  g0.z = (unsigned)(ga & 0xFFFFFFFFull);                // global_addr[31:0]
  g0.w = (unsigned)((ga >> 32) & 0x01FFFFFFull)         // global_addr[56:32]
       | (2u << 30);                                    // type = 2 ("image")
  v8i g1;
  g1[0] = (int)(2u << 16);                              // workgroup_mask=0, data_size=4B
  g1[1] = (int)((tensor_d0 & 0xFFFFu) << 16);           // tensor_dim0[15:0] @ bits 63:48
  g1[2] = (int)((tensor_d0 >> 16) | ((tensor_d1 & 0xFFFFu) << 16));
  g1[3] = (int)((tensor_d1 >> 16) | ((tile_d0 & 0xFFFFu) << 16));
  g1[4] = (int)(tile_d1 & 0xFFFFu);                     // tile_dim1 (tile_dim2=0)
  g1[5] = (int)(stride0_elems & 0xFFFFFFFFull);         // tensor_dim0_stride[31:0]
  g1[6] = (int)((stride0_elems >> 32) & 0xFFFFull);     // stride[47:32]; dim1_stride=0
  g1[7] = 0;
  v4i z4 = {0, 0, 0, 0};                                // groups 2/3 unused (<=2D)
  v8i z8 = {0, 0, 0, 0, 0, 0, 0, 0};
  __builtin_amdgcn_tensor_load_to_lds(g0, g1, z4, z4, z8, 0);
}
#endif

// ---------------- kernel 1: SSD box decode ----------------
__global__ __launch_bounds__(BLK) void ssd_decode_kernel(const float* __restrict__ pred,
                                                         float* __restrict__ boxes) {
  int i = blockIdx.x * BLK + threadIdx.x;
  if (i >= B_IMG * N_ANCH) return;
  const float* p = pred + (size_t)i * N_CH;
  float l0 = p[21], l1 = p[22], l2 = p[23], l3 = p[24];
  float ax = p[25], ay = p[26], aw = p[27], ah = p[28];
  float v0 = p[29], v1 = p[30], v2 = p[31], v3 = p[32];
  float cx = l0 * v0 * aw + ax;
  float cy = l1 * v1 * ah + ay;
  float w  = expf(l2 * v2) * aw;
  float h  = expf(l3 * v3) * ah;
  float4 bx;
  bx.x = (cx - 0.5f * w) * IMG_SC;   // xmin
  bx.y = (cy - 0.5f * h) * IMG_SC;   // ymin
  bx.z = (cx + 0.5f * w) * IMG_SC;   // xmax
  bx.w = (cy + 0.5f * h) * IMG_SC;   // ymax
  reinterpret_cast<float4*>(boxes)[i] = bx;
}

// ---------------- block argmax with (value, lowest-index) tie-break ----------------
__device__ __forceinline__ void block_argmax(float& bv, int& bi, float* rv, int* ri) {
  int tid = threadIdx.x;
  // wave32 shuffle reduction
  for (int off = 16; off > 0; off >>= 1) {
    float ov = __shfl_down(bv, off, 32);
    int   oi = __shfl_down(bi, off, 32);
    if (ov > bv || (ov == bv && oi < bi)) { bv = ov; bi = oi; }
  }
  __syncthreads();                       // protect rv/ri reuse across calls
  if ((tid & 31) == 0) { rv[tid >> 5] = bv; ri[tid >> 5] = bi; }
  __syncthreads();
  if (tid == 0) {
    float Bv = rv[0]; int Bi = ri[0];
#pragma unroll
    for (int w = 1; w < BLK / 32; ++w) {
      float ov = rv[w]; int oi = ri[w];
      if (ov > Bv || (ov == Bv && oi < Bi)) { Bv = ov; Bi = oi; }
    }
    rv[0] = Bv; ri[0] = Bi;
  }
  __syncthreads();
  bv = rv[0]; bi = ri[0];
}

// ---------------- kernel 2: greedy NMS, one block per (class, image) ----------------
// Boxes live in LDS (139.7KB, fits CDNA5's 320KB/WGP); per-class scores live in
// registers (KPT=35 per thread) so suppression touches no shared memory.
__global__ __launch_bounds__(BLK) void ssd_nms_kernel(const float* __restrict__ pred,
                                                      const float* __restrict__ boxes,
                                                      float* __restrict__ rows) {
  const int c   = blockIdx.x;          // 0..19  (class id = c+1)
  const int b   = blockIdx.y;          // image
  const int tid = threadIdx.x;

  extern __shared__ float s_box[];     // N_ANCH*4 floats
  __shared__ float rv[BLK / 32];
  __shared__ int   ri[BLK / 32];

  const float* gbox = boxes + (size_t)b * N_ANCH * 4;
#if HAVE_TDM
  if (tid < 32) {                      // wave 0 issues one TDM DMA (EXEC ignored)
    tdm_load_f32((unsigned)(uintptr_t)(void*)s_box, gbox,
                 (unsigned)(N_ANCH * 4), 1u,
                 (unsigned)(N_ANCH * 4), 1u,
                 (unsigned long long)(N_ANCH * 4));
    __builtin_amdgcn_s_wait_tensorcnt(0);
  }
#else
  for (int i = tid; i < N_ANCH * 4; i += BLK) s_box[i] = gbox[i];
#endif

  // per-class scores -> registers, thresholded (s0 = score>0.5 ? score : NEG)
  float sreg[KPT];
  const float* sp = pred + (size_t)b * N_ANCH * N_CH + (1 + c);
#pragma unroll
  for (int k = 0; k < KPT; ++k) {
    int i = tid + k * BLK;
    if (i < N_ANCH) {
      float v = sp[(size_t)i * N_CH];
      sreg[k] = (v > CONF_TH) ? v : NEGF;
    } else {
      sreg[k] = -__builtin_inff();
    }
  }
  __syncthreads();                     // boxes visible to all waves

  float* rowBase = rows + ((size_t)b * ROWS_PER_IMG + (size_t)c * NMS_MAX) * 6;
  int it = 0;
  for (; it < NMS_MAX; ++it) {
    // ---- argmax over working scores ----
    float bv = -__builtin_inff(); int bi = 0x7FFFFFFF;
#pragma unroll
    for (int k = 0; k < KPT; ++k) {
      int i = tid + k * BLK;
      float v = sreg[k];
      if (v > bv) { bv = v; bi = i; }  // ascending i => lowest index on ties
    }
    block_argmax(bv, bi, rv, ri);

    if (!(bv > NEGF * 0.5f)) break;    // nothing left above threshold (monotone)

    float bx0 = s_box[bi * 4 + 0], by0 = s_box[bi * 4 + 1];
    float bx1 = s_box[bi * 4 + 2], by1 = s_box[bi * 4 + 3];
    if (tid == 0) {                    // emit row (cid, score, box)
      float* r = rowBase + it * 6;
      r[0] = (float)(c + 1); r[1] = bv;
      r[2] = bx0; r[3] = by0; r[4] = bx1; r[5] = by1;
    }

    // ---- suppress IoU > 0.45 (registers only, no barrier needed) ----
    float a1 = fmaxf(bx1 - bx0, 0.f) * fmaxf(by1 - by0, 0.f);
#pragma unroll
    for (int k = 0; k < KPT; ++k) {
      int i = tid + k * BLK;
      if (i < N_ANCH) {
        float x0 = s_box[i * 4 + 0], y0 = s_box[i * 4 + 1];
        float x1 = s_box[i * 4 + 2], y1 = s_box[i * 4 + 3];
        float ix0 = fmaxf(bx0, x0), iy0 = fmaxf(by0, y0);
        float ix1 = fminf(bx1, x1), iy1 = fminf(by1, y1);
        float inter = fmaxf(ix1 - ix0, 0.f) * fmaxf(iy1 - iy0, 0.f);
        float a2 = fmaxf(x1 - x0, 0.f) * fmaxf(y1 - y0, 0.f);
        float iou = inter / fmaxf(a1 + a2 - inter, 1e-8f);
        if (iou > IOU_TH) sreg[k] = NEGF;
      }
    }
  }
  // zero-fill remaining (not-ok) rows, matching rows*okf in the reference
  for (int j = it * 6 + tid; j < NMS_MAX * 6; j += BLK) rowBase[j] = 0.0f;
}

// ---------------- kernel 3: per-image top-200 (stable ties like lax.top_k) ----------------
__global__ __launch_bounds__(BLK) void ssd_topk_kernel(const float* __restrict__ rows,
                                                       float* __restrict__ out) {
  __shared__ float sc[ROWS_PER_IMG];
  __shared__ float rv[BLK / 32];
  __shared__ int   ri[BLK / 32];
  const int b = blockIdx.x, tid = threadIdx.x;
  const float* R = rows + (size_t)b * ROWS_PER_IMG * 6;

  for (int i = tid; i < ROWS_PER_IMG; i += BLK) sc[i] = R[i * 6 + 1];
  __syncthreads();

  for (int k = 0; k < TOP_K; ++k) {
    float bv = -__builtin_inff(); int bi = 0x7FFFFFFF;
    for (int i = tid; i < ROWS_PER_IMG; i += BLK) {
      float v = sc[i];
      if (v > bv) { bv = v; bi = i; }
    }
    block_argmax(bv, bi, rv, ri);
    if (tid < 6) out[((size_t)b * TOP_K + k) * 6 + tid] = R[(size_t)bi * 6 + tid];
    if (tid == 0) sc[bi] = -__builtin_inff();
    __syncthreads();                  // sc update visible before next scan
  }
}

// ---------------- host launcher ----------------
extern "C" void kernel_launch(void* const* d_in, const int* in_sizes, int n_in,
                              void* d_out, int out_size, void* d_ws, size_t ws_size,
                              hipStream_t stream) {
  (void)in_sizes; (void)n_in; (void)out_size; (void)ws_size;
  const float* pred = (const float*)d_in[0];
  float* out = (float*)d_out;

  // workspace: boxes [B,N,4] (8.94MB) then rows [B,2000,6] (3.07MB)
  float* boxes = (float*)d_ws;
  float* rows  = boxes + (size_t)B_IMG * N_ANCH * 4;

  int total = B_IMG * N_ANCH;
  ssd_decode_kernel<<<(total + BLK - 1) / BLK, BLK, 0, stream>>>(pred, boxes);

  size_t smem = (size_t)N_ANCH * 4 * sizeof(float);   // 139,712 B < 320KB/WGP
  (void)hipFuncSetAttribute(reinterpret_cast<const void*>(ssd_nms_kernel),
                            hipFuncAttributeMaxDynamicSharedMemorySize, (int)smem);
  ssd_nms_kernel<<<dim3(N_USED, B_IMG), BLK, smem, stream>>>(pred, boxes, rows);

  ssd_topk_kernel<<<B_IMG, BLK, 0, stream>>>(rows, out);
}